// NodeModel_13984413516158
// MI455X (gfx1250) — compile-verified
//
#include <hip/hip_runtime.h>
#include <math.h>

// ---------------------------------------------------------------------------
// GATv2 layer on MI455X (gfx1250, wave32).
//   FIN = 128, HEADS=4, C=32 (H*C = 128), F_EDGE = 16
// Phases:
//   0) memset accumulators
//   1) proj_kernel: x_l = x@Wl^T+bl, x_r = x@Wr^T+br  via V_WMMA_F32_16X16X4_F32
//   2) deg_ea_kernel / ea_mean_kernel: self-loop edge attr (fill='mean')
//   3) edge_logits_kernel: fused e-projection + leaky_relu + att-dot + seg-max
//   4) exp_kernel: p = exp(l - max), seg-sum denom
//   5) agg_kernel: acc[dst] += alpha * x_l[src]  (hardware f32 atomics)
//   6) final_kernel: out = relu(acc + bias) + x
// ---------------------------------------------------------------------------

typedef __attribute__((ext_vector_type(2))) float v2f;
typedef __attribute__((ext_vector_type(8))) float v8f;

__device__ __forceinline__ void atomicAddF(float* p, float v) {
#ifdef __HIP_PLATFORM_AMD__
  unsafeAtomicAdd(p, v);   // lowers to global_atomic_add_f32
#else
  atomicAdd(p, v);
#endif
}

// float atomic max via monotone integer mapping (init bits must be 0xFFxxxxxx)
__device__ __forceinline__ void atomicMaxFloat(float* addr, float val) {
  if (val >= 0.0f) atomicMax((int*)addr, __float_as_int(val));
  else             atomicMin((unsigned int*)addr, (unsigned int)__float_as_int(val));
}

// ---------------------------------------------------------------------------
// 1) Node projections with f32 WMMA.
//    Block = 256 thr (8 waves) handles 32 rows x 128 cols for BOTH Wl and Wr:
//    32 tiles of 16x16, 4 tiles/wave, K loop 128 in steps of 4 with two
//    independent accumulator chains to break the WMMA RAW dependency.
// ---------------------------------------------------------------------------
__global__ void proj_kernel(const float* __restrict__ x,
                            const float* __restrict__ Wl, const float* __restrict__ bl,
                            const float* __restrict__ Wr, const float* __restrict__ br,
                            float* __restrict__ xl, float* __restrict__ xr, int N) {
  __shared__ float sx[32 * 128];   // 16 KB strip of x
  const int rowbase = blockIdx.x * 32;
  const int t = threadIdx.x;

  // cooperative, coalesced float4 staging of the 32x128 strip
  #pragma unroll
  for (int i = 0; i < 4; ++i) {
    int idx4 = t + i * 256;          // 0..1023 float4s
    int r    = idx4 >> 5;            // 32 float4 per row
    int c4   = idx4 & 31;
    float4 v = make_float4(0.f, 0.f, 0.f, 0.f);
    int row  = rowbase + r;
    if (row < N) v = *(const float4*)(x + (size_t)row * 128 + c4 * 4);
    *(float4*)(sx + r * 128 + c4 * 4) = v;
  }
  __syncthreads();

  const int lane = t & 31;
  const int wave = t >> 5;
  const int hf   = lane >> 4;   // half: K/M offset selector
  const int l16  = lane & 15;
  const bool full = (rowbase + 32 <= N);   // block-uniform fast path

  #pragma unroll
  for (int i = 0; i < 4; ++i) {
    const int T  = wave + 8 * i;       // 0..31
    const int m  = T >> 4;             // 0 -> Wl, 1 -> Wr
    const int rt = (T >> 3) & 1;       // row tile 0/1
    const int ct = T & 7;              // col tile 0..7
    const float* W  = m ? Wr : Wl;
    const float* bs = m ? br : bl;
    float*       Y  = m ? xr : xl;

    v8f acc0 = {}, acc1 = {};
    // A frag: lane holds X[rowtile + l16][k + 2*hf + {0,1}]   (ISA 16x4 layout)
    const float* arow = sx + (size_t)(rt * 16 + l16) * 128 + 2 * hf;
    // B frag: lane holds W[ct*16 + l16][k + 2*hf + {0,1}]  (B = W^T tile, KxN)
    const float* brow = W + (size_t)(ct * 16 + l16) * 128 + 2 * hf;

    #pragma unroll 2
    for (int k = 0; k < 128; k += 8) {
      v2f a0 = *(const v2f*)(arow + k);
      v2f b0 = *(const v2f*)(brow + k);
      acc0 = __builtin_amdgcn_wmma_f32_16x16x4_f32(false, a0, false, b0,
                                                   (short)0, acc0, false, false);
      v2f a1 = *(const v2f*)(arow + k + 4);
      v2f b1 = *(const v2f*)(brow + k + 4);
      acc1 = __builtin_amdgcn_wmma_f32_16x16x4_f32(false, a1, false, b1,
                                                   (short)0, acc1, false, false);
    }
    v8f acc = acc0 + acc1;

    const int col = ct * 16 + l16;
    const float bv = bs[col];
    float* ybase = Y + (size_t)(rowbase + rt * 16) * 128 + col;
    if (full) {
      // straight-line stores, no exec-mask churn (C/D layout: M = r + 8*half)
      #pragma unroll
      for (int r = 0; r < 8; ++r)
        ybase[(size_t)(r + 8 * hf) * 128] = acc[r] + bv;
    } else {
      #pragma unroll
      for (int r = 0; r < 8; ++r) {
        int row = rowbase + rt * 16 + r + 8 * hf;
        if (row < N) Y[(size_t)row * 128 + col] = acc[r] + bv;
      }
    }
  }
}

// ---------------------------------------------------------------------------
// 2) degree + edge_attr sums per destination (thread per (edge, feature))
// ---------------------------------------------------------------------------
__global__ void deg_ea_kernel(const int* __restrict__ dst, const float* __restrict__ ea,
                              float* __restrict__ deg, float* __restrict__ easum, int E) {
  int idx = blockIdx.x * blockDim.x + threadIdx.x;
  if (idx >= E * 16) return;
  int e = idx >> 4, f = idx & 15;
  int d = dst[e];
  atomicAddF(easum + (size_t)d * 16 + f, ea[idx]);
  if (f == 0) atomicAddF(deg + d, 1.0f);
}

__global__ void ea_mean_kernel(float* __restrict__ easum, const float* __restrict__ deg, int N) {
  int idx = blockIdx.x * blockDim.x + threadIdx.x;
  if (idx >= N * 16) return;
  easum[idx] = easum[idx] / fmaxf(deg[idx >> 4], 1.0f);  // in-place -> ea_mean
}

// ---------------------------------------------------------------------------
// 3) Fused logits: wave per edge, 4 channels/lane.
//    e = We @ ea computed in-register (We row = 16 contiguous floats).
// ---------------------------------------------------------------------------
__global__ void edge_logits_kernel(const int* __restrict__ src, const int* __restrict__ dst,
                                   const float* __restrict__ ea, const float* __restrict__ eamean,
                                   const float* __restrict__ xl, const float* __restrict__ xr,
                                   const float* __restrict__ We, const float* __restrict__ att,
                                   float* __restrict__ logits, float* __restrict__ maxb,
                                   int E, int N) {
  int wid  = (blockIdx.x * blockDim.x + threadIdx.x) >> 5;
  int lane = threadIdx.x & 31;
  if (wid >= E + N) return;

  int s, d; const float* eap;
  if (wid < E) { s = src[wid]; d = dst[wid]; eap = ea + (size_t)wid * 16; }
  else         { s = d = wid - E;            eap = eamean + (size_t)s * 16; }

  const int ch = lane * 4;                               // 4 channels per lane
  const float4 vl = *(const float4*)(xl + (size_t)s * 128 + ch);
  const float4 vr = *(const float4*)(xr + (size_t)d * 128 + ch);

  float eav[16];
  #pragma unroll
  for (int f = 0; f < 16; ++f) eav[f] = eap[f];          // uniform -> broadcast

  const float xlv[4] = {vl.x, vl.y, vl.z, vl.w};
  const float xrv[4] = {vr.x, vr.y, vr.z, vr.w};
  float partial = 0.0f;
  #pragma unroll
  for (int j = 0; j < 4; ++j) {
    const float* wrow = We + (size_t)(ch + j) * 16;
    float ev = 0.0f;
    #pragma unroll
    for (int f = 0; f < 16; ++f) ev = fmaf(wrow[f], eav[f], ev);
    float a = xlv[j] + xrv[j] + ev;
    a = (a > 0.0f) ? a : 0.2f * a;                       // leaky_relu
    partial = fmaf(a, att[ch + j], partial);             // att flat == channel idx
  }
  // reduce within aligned groups of 8 lanes (one head per group)
  partial += __shfl_xor(partial, 1, 32);
  partial += __shfl_xor(partial, 2, 32);
  partial += __shfl_xor(partial, 4, 32);

  if ((lane & 7) == 0) {
    int h = lane >> 3;
    logits[(size_t)wid * 4 + h] = partial;
    atomicMaxFloat(maxb + (size_t)d * 4 + h, partial);
  }
}

// ---------------------------------------------------------------------------
// 4) p = exp(l - segmax); denom[dst] += p      (thread per (edge, head))
// ---------------------------------------------------------------------------
__global__ void exp_kernel(const int* __restrict__ dst, float* __restrict__ lp,
                           const float* __restrict__ maxb, float* __restrict__ denom,
                           int E, int N) {
  int idx = blockIdx.x * blockDim.x + threadIdx.x;
  if (idx >= (E + N) * 4) return;
  int e = idx >> 2, h = idx & 3;
  int d = (e < E) ? dst[e] : e - E;
  float p = __expf(lp[idx] - maxb[(size_t)d * 4 + h]);
  lp[idx] = p;
  atomicAddF(denom + (size_t)d * 4 + h, p);
}

// ---------------------------------------------------------------------------
// 5) acc[dst] += alpha * x_l[src]   (wave per edge, 4 ch/lane, f32 atomics)
// ---------------------------------------------------------------------------
__global__ void agg_kernel(const int* __restrict__ src, const int* __restrict__ dst,
                           const float* __restrict__ lp, const float* __restrict__ denom,
                           const float* __restrict__ xl, float* __restrict__ acc,
                           int E, int N) {
  int wid  = (blockIdx.x * blockDim.x + threadIdx.x) >> 5;
  int lane = threadIdx.x & 31;
  if (wid >= E + N) return;
  int s, d;
  if (wid < E) { s = src[wid]; d = dst[wid]; } else { s = d = wid - E; }
  int h = lane >> 3;
  float alpha = lp[(size_t)wid * 4 + h] / denom[(size_t)d * 4 + h];
  int ch = lane * 4;
  float4 v = *(const float4*)(xl + (size_t)s * 128 + ch);
  float* ap = acc + (size_t)d * 128 + ch;
  atomicAddF(ap + 0, alpha * v.x);
  atomicAddF(ap + 1, alpha * v.y);
  atomicAddF(ap + 2, alpha * v.z);
  atomicAddF(ap + 3, alpha * v.w);
}

// ---------------------------------------------------------------------------
// 6) out = relu(acc + bias) + x
// ---------------------------------------------------------------------------
__global__ void final_kernel(const float* __restrict__ acc, const float* __restrict__ bias,
                             const float* __restrict__ x, float* __restrict__ out, int N) {
  int idx = blockIdx.x * blockDim.x + threadIdx.x;
  if (idx >= N * 128) return;
  float v = acc[idx] + bias[idx & 127];
  out[idx] = fmaxf(v, 0.0f) + x[idx];
}

// ---------------------------------------------------------------------------
extern "C" void kernel_launch(void* const* d_in, const int* in_sizes, int n_in,
                              void* d_out, int out_size, void* d_ws, size_t ws_size,
                              hipStream_t stream) {
  const float* x    = (const float*)d_in[0];
  const int*   ei   = (const int*)  d_in[1];
  const float* ea   = (const float*)d_in[2];
  // d_in[3] (u) and d_in[4] (batch) unused by the reference node path
  const float* Wl   = (const float*)d_in[5];
  const float* bl   = (const float*)d_in[6];
  const float* Wr   = (const float*)d_in[7];
  const float* br   = (const float*)d_in[8];
  const float* We   = (const float*)d_in[9];
  const float* att  = (const float*)d_in[10];
  const float* bias = (const float*)d_in[11];
  float* out = (float*)d_out;

  const int N = in_sizes[0] / 128;
  const int E = in_sizes[1] / 2;
  const int* src = ei;
  const int* dst = ei + E;

  // workspace partition (floats)
  float* ws    = (float*)d_ws;
  float* xl    = ws;
  float* xr    = xl    + (size_t)N * 128;
  float* acc   = xr    + (size_t)N * 128;
  float* easum = acc   + (size_t)N * 128;   // becomes ea_mean in-place
  float* deg   = easum + (size_t)N * 16;
  float* maxb  = deg   + (size_t)N;
  float* denom = maxb  + (size_t)N * 4;
  float* lp    = denom + (size_t)N * 4;     // logits -> p, (E+N)*4

  // per-call re-init (graph-capture-safe)
  hipMemsetAsync(acc,   0,    (size_t)N * 128 * sizeof(float), stream);
  hipMemsetAsync(easum, 0,    (size_t)N * 16  * sizeof(float), stream);
  hipMemsetAsync(deg,   0,    (size_t)N       * sizeof(float), stream);
  hipMemsetAsync(denom, 0,    (size_t)N * 4   * sizeof(float), stream);
  hipMemsetAsync(maxb,  0xFF, (size_t)N * 4   * sizeof(float), stream); // -NaN bits: valid -inf init for int-trick max

  const int TB = 256;
  // 1) projections (WMMA)
  proj_kernel<<<(N + 31) / 32, TB, 0, stream>>>(x, Wl, bl, Wr, br, xl, xr, N);
  // 2) self-loop mean edge attr
  deg_ea_kernel<<<((size_t)E * 16 + TB - 1) / TB, TB, 0, stream>>>(dst, ea, deg, easum, E);
  ea_mean_kernel<<<((size_t)N * 16 + TB - 1) / TB, TB, 0, stream>>>(easum, deg, N);
  // 3) fused logits + segment max  (wave per edge)
  const size_t waves = (size_t)E + N;
  edge_logits_kernel<<<(waves * 32 + TB - 1) / TB, TB, 0, stream>>>(
      src, dst, ea, easum, xl, xr, We, att, lp, maxb, E, N);
  // 4) exp + segment sum
  exp_kernel<<<(waves * 4 + TB - 1) / TB, TB, 0, stream>>>(dst, lp, maxb, denom, E, N);
  // 5) weighted aggregation
  agg_kernel<<<(waves * 32 + TB - 1) / TB, TB, 0, stream>>>(src, dst, lp, denom, xl, acc, E, N);
  // 6) epilogue
  final_kernel<<<((size_t)N * 128 + TB - 1) / TB, TB, 0, stream>>>(acc, bias, x, out, N);
}